// GraphSAGELinkPredictor_21131239096608
// MI455X (gfx1250) — compile-verified
//
#include <hip/hip_runtime.h>
#include <hip/hip_bf16.h>

// GraphSAGE link predictor for MI455X (gfx1250, wave32).
// fp32 throughout (workload is L2/atomic bound, not compute bound);
// GEMMs use V_WMMA_F32_16X16X4_F32 (fp32 in / fp32 accumulate).

typedef __attribute__((ext_vector_type(2))) float v2f;
typedef __attribute__((ext_vector_type(8))) float v8f;

#define D 128  // D_IN == D_HID == 128

// ---------------------------------------------------------------- zero
__global__ void zero_kernel(float* __restrict__ p, size_t n) {
  size_t i = (size_t)blockIdx.x * blockDim.x + threadIdx.x;
  size_t stride = (size_t)gridDim.x * blockDim.x;
  for (; i < n; i += stride) p[i] = 0.0f;
}

// ---------------------------------------------------------------- scatter-sum
// One wave32 per edge: lane l moves floats [4l, 4l+4) of the 128-wide row.
__global__ void scatter_kernel(const float* __restrict__ feat,
                               const int* __restrict__ src,
                               const int* __restrict__ dst,
                               float* __restrict__ agg,
                               float* __restrict__ cnt,
                               int n_edges) {
  int e = (int)(((size_t)blockIdx.x * blockDim.x + threadIdx.x) >> 5);
  int lane = threadIdx.x & 31;
  if (e >= n_edges) return;
  int s = src[e];
  int d = dst[e];
  const float4 v = *(const float4*)(feat + (size_t)s * D + lane * 4);
  float* ap = agg + (size_t)d * D + lane * 4;
  atomicAdd(ap + 0, v.x);
  atomicAdd(ap + 1, v.y);
  atomicAdd(ap + 2, v.z);
  atomicAdd(ap + 3, v.w);
  if (lane == 0) atomicAdd(cnt + d, 1.0f);
}

// ---------------------------------------------------------------- fused SAGE GEMM
// out = maybe_relu( (agg / max(cnt,1)) @ Wl + xin @ Wr + bias )
// Block: 128 threads = 4 waves; wave w handles M-tile (blockIdx.y*4 + w),
// blockIdx.x selects the 16-wide N-tile. Wl/Wr column slices staged in LDS.
//
// V_WMMA_F32_16X16X4_F32 VGPR layout (ISA 7.12.2, 32-bit tables):
//   A 16x4:  lanes 0-15 -> M=lane, {v0,v1}={K+0,K+1}; lanes 16-31 -> {K+2,K+3}
//   B 4x16:  lanes 0-15 -> N=lane, {v0,v1}=rows {K+0,K+1}; lanes 16-31 -> {K+2,K+3}
//   C 16x16: VGPR j: lanes 0-15 -> M=j, lanes 16-31 -> M=j+8, N=lane&15
__global__ void sage_gemm_kernel(const float* __restrict__ agg,
                                 const float* __restrict__ cnt,
                                 const float* __restrict__ xin,
                                 const float* __restrict__ Wl,
                                 const float* __restrict__ Wr,
                                 const float* __restrict__ bias,
                                 float* __restrict__ out,
                                 int n_rows, int do_relu) {
  __shared__ float lWl[D * 16];
  __shared__ float lWr[D * 16];

  const int tid = threadIdx.x;   // 0..127
  const int wave = tid >> 5;     // 0..3
  const int lane = tid & 31;
  const int n0 = blockIdx.x * 16;

  // Stage 128x16 column slices of Wl and Wr into LDS (float4 granularity).
  for (int i4 = tid; i4 < 512; i4 += 128) {
    int k = i4 >> 2;
    int c = (i4 & 3) << 2;
    *(float4*)(&lWl[k * 16 + c]) = *(const float4*)(&Wl[(size_t)k * D + n0 + c]);
    *(float4*)(&lWr[k * 16 + c]) = *(const float4*)(&Wr[(size_t)k * D + n0 + c]);
  }
  __syncthreads();

  const int mtile = blockIdx.y * 4 + wave;
  if (mtile * 16 >= n_rows) return;  // wave-uniform: EXEC stays all-ones for WMMA

  const int m0 = mtile * 16;
  const int lrow = lane & 15;   // A-row within tile / B,C column
  const int khalf = lane >> 4;  // 0: K+{0,1}, 1: K+{2,3}
  const int row = m0 + lrow;

  const float inv = 1.0f / fmaxf(cnt[row], 1.0f);  // scatter-mean normalization

  const float* aggRow = agg + (size_t)row * D + 2 * khalf;
  const float* xRow = xin + (size_t)row * D + 2 * khalf;

  v8f acc = {};
#pragma unroll 4
  for (int k = 0; k < D; k += 4) {
    v2f a0, b0, a1, b1v;
    a0.x = aggRow[k] * inv;
    a0.y = aggRow[k + 1] * inv;
    b0.x = lWl[(k + 2 * khalf) * 16 + lrow];
    b0.y = lWl[(k + 2 * khalf + 1) * 16 + lrow];
    acc = __builtin_amdgcn_wmma_f32_16x16x4_f32(false, a0, false, b0,
                                                (short)0, acc, false, false);
    a1.x = xRow[k];
    a1.y = xRow[k + 1];
    b1v.x = lWr[(k + 2 * khalf) * 16 + lrow];
    b1v.y = lWr[(k + 2 * khalf + 1) * 16 + lrow];
    acc = __builtin_amdgcn_wmma_f32_16x16x4_f32(false, a1, false, b1v,
                                                (short)0, acc, false, false);
  }

  const float bv = bias[n0 + lrow];
#pragma unroll
  for (int j = 0; j < 8; ++j) {
    float v = acc[j] + bv;
    if (do_relu) v = fmaxf(v, 0.0f);
    out[(size_t)(m0 + j + 8 * khalf) * D + n0 + lrow] = v;
  }
}

// ---------------------------------------------------------------- link decode
// score = concat(z[src], z[dst]) . Wp + bp ; one wave32 per predicted edge.
__global__ void decode_kernel(const float* __restrict__ z,
                              const int* __restrict__ ei, int n_pred,
                              const float* __restrict__ Wp,
                              const float* __restrict__ bp,
                              float* __restrict__ out) {
  int e = (int)(((size_t)blockIdx.x * blockDim.x + threadIdx.x) >> 5);
  int lane = threadIdx.x & 31;
  if (e >= n_pred) return;
  int s = ei[e];
  int d = ei[n_pred + e];
  float4 zs = *(const float4*)(z + (size_t)s * D + lane * 4);
  float4 zd = *(const float4*)(z + (size_t)d * D + lane * 4);
  float4 ws = *(const float4*)(Wp + lane * 4);
  float4 wd = *(const float4*)(Wp + D + lane * 4);
  float p = zs.x * ws.x + zs.y * ws.y + zs.z * ws.z + zs.w * ws.w +
            zd.x * wd.x + zd.y * wd.y + zd.z * wd.z + zd.w * wd.w;
#pragma unroll
  for (int off = 16; off > 0; off >>= 1) p += __shfl_xor(p, off, 32);
  if (lane == 0) out[e] = p + bp[0];
}

// ---------------------------------------------------------------- launch
extern "C" void kernel_launch(void* const* d_in, const int* in_sizes, int n_in,
                              void* d_out, int out_size, void* d_ws, size_t ws_size,
                              hipStream_t stream) {
  const float* x = (const float*)d_in[0];
  const int* edge_index = (const int*)d_in[1];
  // d_in[2] = edge_weight: unused by the reference
  const int* pos_ei = (const int*)d_in[3];
  const int* neg_ei = (const int*)d_in[4];
  const float* W1l = (const float*)d_in[5];
  const float* W1r = (const float*)d_in[6];
  const float* b1 = (const float*)d_in[7];
  const float* W2l = (const float*)d_in[8];
  const float* W2r = (const float*)d_in[9];
  const float* b2 = (const float*)d_in[10];
  const float* Wp = (const float*)d_in[11];
  const float* bp = (const float*)d_in[12];
  float* out = (float*)d_out;

  const int n_nodes = in_sizes[0] / D;      // 50000
  const int n_edges = in_sizes[1] / 2;      // 800000
  const int n_pred = in_sizes[3] / 2;       // 100000

  const size_t feat = (size_t)n_nodes * D;
  float* ws = (float*)d_ws;
  float* agg = ws;               // [n_nodes, D]
  float* cnt = agg + feat;       // [n_nodes]
  float* h   = cnt + n_nodes;    // [n_nodes, D]
  float* z   = h + feat;         // [n_nodes, D]

  const int scat_blocks = (int)(((size_t)n_edges * 32 + 255) / 256);
  const dim3 gemm_grid(D / 16, (n_nodes / 16 + 3) / 4);
  const int dec_blocks = (int)(((size_t)n_pred * 32 + 255) / 256);

  // ---- layer 1: h = relu(mean-agg(x) @ W1l + x @ W1r + b1)
  zero_kernel<<<1024, 256, 0, stream>>>(agg, feat + n_nodes);
  scatter_kernel<<<scat_blocks, 256, 0, stream>>>(x, edge_index,
                                                  edge_index + n_edges,
                                                  agg, cnt, n_edges);
  sage_gemm_kernel<<<gemm_grid, 128, 0, stream>>>(agg, cnt, x, W1l, W1r, b1,
                                                  h, n_nodes, 1);

  // ---- layer 2: z = mean-agg(h) @ W2l + h @ W2r + b2
  zero_kernel<<<1024, 256, 0, stream>>>(agg, feat + n_nodes);
  scatter_kernel<<<scat_blocks, 256, 0, stream>>>(h, edge_index,
                                                  edge_index + n_edges,
                                                  agg, cnt, n_edges);
  sage_gemm_kernel<<<gemm_grid, 128, 0, stream>>>(agg, cnt, h, W2l, W2r, b2,
                                                  z, n_nodes, 0);

  // ---- decode pos / neg
  decode_kernel<<<dec_blocks, 256, 0, stream>>>(z, pos_ei, n_pred, Wp, bp, out);
  decode_kernel<<<dec_blocks, 256, 0, stream>>>(z, neg_ei, n_pred, Wp, bp,
                                                out + n_pred);
}